// BatchGeomLoss_48679159333130
// MI455X (gfx1250) — compile-verified
//
#include <hip/hip_runtime.h>

#define GRAPHS 64
#define NPTS   1024
#define DIM    16
#define STRIDE 18          // padded LDS row stride (dwords): 18*m % 64 distinct for m=0..15
#define N_ITERS 20
#define EPSV    0.0025f    // blur^2
#define INV_EPS 400.0f
#define LOG2E   1.44269504088896f
#define LN2     0.69314718055995f
#define K2      (INV_EPS * LOG2E)      // Gram-matrix scale into log2 domain
#define LOGW2   (-10.0f)               // log2(1/1024): weights term in log2 domain

typedef __attribute__((ext_vector_type(2))) float v2f;
typedef __attribute__((ext_vector_type(8))) float v8f;

// LDS (address_space(3)) typed views — 32-bit ds addressing, no generic-pointer checks.
typedef __attribute__((address_space(3))) float lfloat;
typedef __attribute__((address_space(3))) v2f   lv2f;

// dynamic-LDS layout (float offsets)
#define OFF_X   0
#define OFF_Y   (NPTS*STRIDE)
#define OFF_HX  (2*NPTS*STRIDE)
#define OFF_HY  (OFF_HX + NPTS)
#define OFF_F   (OFF_HY + NPTS)
#define OFF_G   (OFF_F + NPTS)
#define OFF_GY  (OFF_G + NPTS)
#define OFF_RED (OFF_GY + NPTS)
#define SMEM_FLOATS (OFF_RED + 8)

__device__ __forceinline__ float exp2_raw(float x) { return __builtin_amdgcn_exp2f(x); }
__device__ __forceinline__ float log2_raw(float x) { return __builtin_amdgcn_logf(x); }

__device__ __forceinline__ void lse2_combine(float& m, float& s, float om, float os) {
    float nm = fmaxf(m, om);
    s = s * exp2_raw(m - nm) + os * exp2_raw(om - nm);
    m = nm;
}

// One log2-domain logsumexp sweep:
//   Fout[i] = -eps*ln2 * log2( sum_j 2^( P_i·Q_j*K2 + Gcol[j] - Hrow[i] ) )
// (Hrow/Gcol are pre-scaled into log2 units; Fout is in natural units.)
__device__ __forceinline__ void sinkhorn_sweep(const lfloat* Pl,
                                               const lfloat* Ql,
                                               const lfloat* Hrow,
                                               const lfloat* Gcol,
                                               lfloat*       Fout,
                                               int wave, int lane)
{
    const int m16 = lane & 15;
    const int hi  = lane >> 4;

    for (int rb = wave; rb < NPTS / 16; rb += 8) {
        const int rowbase = rb * 16;

        // A fragments (16x4 f32, K accumulated in 4 steps) — invariant over column loop
        v2f a[4];
#pragma unroll
        for (int k0 = 0; k0 < 4; ++k0)
            a[k0] = *(const lv2f*)&Pl[(rowbase + m16) * STRIDE + 4 * k0 + 2 * hi];

        float hxv[8], mAcc[8], sAcc[8];
#pragma unroll
        for (int v = 0; v < 8; ++v) {
            hxv[v]  = Hrow[rowbase + 8 * hi + v];
            mAcc[v] = -1e30f;
            sAcc[v] = 0.0f;
        }

#pragma unroll 2
        for (int cb = 0; cb < NPTS / 16; ++cb) {
            const int colbase = cb * 16;
            v2f b[4];
#pragma unroll
            for (int k0 = 0; k0 < 4; ++k0)
                b[k0] = *(const lv2f*)&Ql[(colbase + m16) * STRIDE + 4 * k0 + 2 * hi];

            v8f d = {};
#pragma unroll
            for (int k0 = 0; k0 < 4; ++k0)
                d = __builtin_amdgcn_wmma_f32_16x16x4_f32(false, a[k0], false, b[k0],
                                                          (short)0, d, false, false);

            const float gyc = Gcol[colbase + m16];
#pragma unroll
            for (int v = 0; v < 8; ++v) {
                float val = fmaf(d[v], K2, gyc - hxv[v]);   // log2-domain score
                float nm  = fmaxf(mAcc[v], val);
                sAcc[v]   = sAcc[v] * exp2_raw(mAcc[v] - nm) + exp2_raw(val - nm);
                mAcc[v]   = nm;
            }
        }

        // merge partial logsumexps across the 16 lanes of each half-wave
#pragma unroll
        for (int off = 1; off < 16; off <<= 1) {
#pragma unroll
            for (int v = 0; v < 8; ++v) {
                float om = __shfl_xor(mAcc[v], off, 32);
                float os = __shfl_xor(sAcc[v], off, 32);
                lse2_combine(mAcc[v], sAcc[v], om, os);
            }
        }
        if (m16 == 0) {
#pragma unroll
            for (int v = 0; v < 8; ++v)
                Fout[rowbase + 8 * hi + v] =
                    (-EPSV * LN2) * (mAcc[v] + log2_raw(sAcc[v]));
        }
    }
}

__global__ void __launch_bounds__(256)
BatchGeomLoss_sinkhorn_kernel(const float* __restrict__ X,
                              const float* __restrict__ Yt,
                              float* __restrict__ out)
{
    extern __shared__ float smem_raw[];
    lfloat* smem = (lfloat*)smem_raw;       // single addrspacecast, folded by backend
    lfloat* Xl  = smem + OFF_X;
    lfloat* Yl  = smem + OFF_Y;
    lfloat* HX  = smem + OFF_HX;
    lfloat* HY  = smem + OFF_HY;
    lfloat* F   = smem + OFF_F;
    lfloat* G   = smem + OFF_G;
    lfloat* GY  = smem + OFF_GY;
    lfloat* RED = smem + OFF_RED;

    const int tid  = threadIdx.x;
    const int wave = tid >> 5;
    const int lane = tid & 31;
    const int wg    = blockIdx.x;
    const int graph = wg / 3;
    const int t     = wg % 3;          // 0: OT(x,y)  1: OT(x,x)  2: OT(y,y)

    const float* Pa = (t == 2) ? (Yt + (size_t)graph * NPTS * DIM)
                               : (X  + (size_t)graph * NPTS * DIM);
    const float* Pb = (t == 0) ? (Yt + (size_t)graph * NPTS * DIM) : Pa;

    // stage both point sets into padded LDS (float4 global loads, 8 MB total HBM)
    for (int i = tid; i < NPTS * DIM / 4; i += 256) {
        const float4 vx = ((const float4*)Pa)[i];
        const float4 vy = ((const float4*)Pb)[i];
        const int row = i >> 2;
        const int c4  = (i & 3) * 4;
        lfloat* xd = &Xl[row * STRIDE + c4];
        xd[0] = vx.x; xd[1] = vx.y; xd[2] = vx.z; xd[3] = vx.w;
        lfloat* yd = &Yl[row * STRIDE + c4];
        yd[0] = vy.x; yd[1] = vy.y; yd[2] = vy.z; yd[3] = vy.w;
    }
    __syncthreads();

    // squared half-norms in log2 units (0.5*||.||^2 / eps * log2e); init g = 0
    for (int i = tid; i < NPTS; i += 256) {
        float sx = 0.f, sy = 0.f;
#pragma unroll
        for (int k = 0; k < DIM; ++k) {
            float u = Xl[i * STRIDE + k]; sx = fmaf(u, u, sx);
            float w = Yl[i * STRIDE + k]; sy = fmaf(w, w, sy);
        }
        HX[i] = 0.5f * sx * K2;
        HY[i] = 0.5f * sy * K2;
        G[i]  = 0.0f;
    }
    __syncthreads();

    // 20 full Sinkhorn iterations + one final f-sweep (41 sweeps)
    for (int it = 0; it <= N_ITERS; ++it) {
        for (int i = tid; i < NPTS; i += 256)
            GY[i] = fmaf(G[i], K2, LOGW2 - HY[i]);   // column term, log2 units
        __syncthreads();
        sinkhorn_sweep(Xl, Yl, HX, GY, F, wave, lane);
        __syncthreads();
        if (it == N_ITERS) break;
        for (int i = tid; i < NPTS; i += 256)
            GY[i] = fmaf(F[i], K2, LOGW2 - HX[i]);
        __syncthreads();
        sinkhorn_sweep(Yl, Xl, HY, GY, G, wave, lane);
        __syncthreads();
    }

    // loss = mean(f) + mean(g); weighted (-0.5 for self terms), batch-mean over 64
    float part = 0.0f;
    for (int i = tid; i < NPTS; i += 256) part += F[i] + G[i];
#pragma unroll
    for (int off = 16; off >= 1; off >>= 1) part += __shfl_xor(part, off, 32);
    if (lane == 0) RED[wave] = part;
    __syncthreads();
    if (tid == 0) {
        float s = 0.f;
#pragma unroll
        for (int w = 0; w < 8; ++w) s += RED[w];
        const float wgt = (t == 0) ? 1.0f : -0.5f;
        atomicAdd(out, wgt * s * (1.0f / (float)(NPTS * GRAPHS)));
    }
}

extern "C" void kernel_launch(void* const* d_in, const int* in_sizes, int n_in,
                              void* d_out, int out_size, void* d_ws, size_t ws_size,
                              hipStream_t stream) {
    const float* x   = (const float*)d_in[0];
    const float* y   = (const float*)d_in[1];
    float*       out = (float*)d_out;

    static_assert(SMEM_FLOATS * sizeof(float) < 320 * 1024, "fits WGP LDS");
    (void)hipFuncSetAttribute((const void*)BatchGeomLoss_sinkhorn_kernel,
                              hipFuncAttributeMaxDynamicSharedMemorySize,
                              SMEM_FLOATS * (int)sizeof(float));

    hipMemsetAsync(out, 0, sizeof(float), stream);
    BatchGeomLoss_sinkhorn_kernel<<<dim3(GRAPHS * 3), dim3(256),
                                    SMEM_FLOATS * sizeof(float), stream>>>(x, y, out);
}